// AttributeDecoder_22385369547414
// MI455X (gfx1250) — compile-verified
//
#include <hip/hip_runtime.h>

typedef __attribute__((ext_vector_type(2))) float v2f;
typedef __attribute__((ext_vector_type(8))) float v8f;

// ---------------------------------------------------------------------------
// Utility: zero a u32 region (graph-capture-safe replacement for memset)
// ---------------------------------------------------------------------------
__global__ void gcn_zero_u32(unsigned* __restrict__ p, long long n) {
    long long i = (long long)blockIdx.x * blockDim.x + threadIdx.x;
    if (i < n) p[i] = 0u;
}

// ---------------------------------------------------------------------------
// Degree: deg[d] += 1 for each edge destination (exact integer atomics)
// ---------------------------------------------------------------------------
__global__ void gcn_degree(const long long* __restrict__ dst,
                           int* __restrict__ deg, int nEdges) {
    int e = blockIdx.x * blockDim.x + threadIdx.x;
    if (e < nEdges) atomicAdd(&deg[(int)dst[e]], 1);
}

// dis[i] = rsqrt(1 + deg[i])
__global__ void gcn_rsqrt(const int* __restrict__ deg,
                          float* __restrict__ dis, int n) {
    int i = blockIdx.x * blockDim.x + threadIdx.x;
    if (i < n) dis[i] = rsqrtf(1.0f + (float)deg[i]);
}

// ---------------------------------------------------------------------------
// GEMM: H[N,Fout] = X[N,Fin] @ W[Fin,Fout] via V_WMMA_F32_16X16X4_F32.
// One wave computes one 16x16 output tile; K advances 4 per WMMA.
// A frag (16x4 f32): lanes 0-15 hold M=lane, K={k,k+1}; lanes 16-31 K={k+2,k+3}.
// B frag (4x16 f32): lanes 0-15 hold N=lane, K={k,k+1}; lanes 16-31 K={k+2,k+3}.
// C/D (16x16 f32):   VGPR j -> row j (lanes 0-15) / row j+8 (lanes 16-31).
// ---------------------------------------------------------------------------
__global__ void gcn_gemm_wmma(const float* __restrict__ X,
                              const float* __restrict__ W,
                              float* __restrict__ H,
                              int nRows, int Fin, int Fout) {
    const int lane = threadIdx.x & 31;
    const int wave = threadIdx.x >> 5;
    const int wavesPerBlock = blockDim.x >> 5;
    const int colTiles = Fout >> 4;
    const int rowTiles = nRows >> 4;           // nRows is a multiple of 16
    int tile = blockIdx.x * wavesPerBlock + wave;  // wave-uniform
    if (tile >= rowTiles * colTiles) return;       // wave-uniform exit
    const int row0 = (tile / colTiles) << 4;
    const int col0 = (tile % colTiles) << 4;

    const int l15   = lane & 15;
    const int khalf = (lane >> 4) << 1;        // 0 for lanes 0-15, 2 for 16-31

    v8f c = {};
    const float* xrow = X + (size_t)(row0 + l15) * Fin;
    for (int k = 0; k < Fin; k += 4) {
        v2f a, b;
        a.x = xrow[k + khalf];
        a.y = xrow[k + khalf + 1];
        b.x = W[(size_t)(k + khalf)     * Fout + col0 + l15];
        b.y = W[(size_t)(k + khalf + 1) * Fout + col0 + l15];
        c = __builtin_amdgcn_wmma_f32_16x16x4_f32(
                /*neg_a=*/false, a, /*neg_b=*/false, b,
                /*c_mod=*/(short)0, c, /*reuse_a=*/false, /*reuse_b=*/false);
    }

    const int mbase = (lane >> 4) << 3;        // 0 or 8
#pragma unroll
    for (int j = 0; j < 8; ++j) {
        H[(size_t)(row0 + mbase + j) * Fout + col0 + l15] = c[j];
    }
}

// ---------------------------------------------------------------------------
// Edge scatter: agg[dst] += h[src] * dis[src]*dis[dst].
// One thread per (edge, 4-float chunk); hardware fp32 global atomics.
// All of h/agg live in the 192MB L2, so this is an L2-atomic-rate problem ->
// flood it with 51M independent wave32 threads.
// ---------------------------------------------------------------------------
__global__ void gcn_scatter(const float* __restrict__ H,
                            const long long* __restrict__ src,
                            const long long* __restrict__ dst,
                            const float* __restrict__ dis,
                            float* __restrict__ agg,
                            long long nWork, int F, int chunkShift) {
    long long tid = (long long)blockIdx.x * blockDim.x + threadIdx.x;
    if (tid >= nWork) return;
    int e = (int)(tid >> chunkShift);
    int c = (int)(tid & ((1 << chunkShift) - 1));
    int s = (int)src[e];
    int d = (int)dst[e];
    float norm = dis[s] * dis[d];
    const float4 hv = *(const float4*)(H + (size_t)s * F + (c << 2));
    float* ap = agg + (size_t)d * F + (c << 2);
    unsafeAtomicAdd(ap + 0, hv.x * norm);
    unsafeAtomicAdd(ap + 1, hv.y * norm);
    unsafeAtomicAdd(ap + 2, hv.z * norm);
    unsafeAtomicAdd(ap + 3, hv.w * norm);
}

// ---------------------------------------------------------------------------
// Finish: out = relu(agg + h * dis^2 + bias)   (elementwise; out may == agg)
// ---------------------------------------------------------------------------
__global__ void gcn_finish(const float* __restrict__ agg,
                           const float* __restrict__ H,
                           const float* __restrict__ dis,
                           const float* __restrict__ bias,
                           float* __restrict__ out,
                           long long nElems, int F, int fShift) {
    long long i = (long long)blockIdx.x * blockDim.x + threadIdx.x;
    if (i >= nElems) return;
    int node = (int)(i >> fShift);
    int f    = (int)(i & (F - 1));
    float ds = dis[node];
    float v  = agg[i] + H[i] * ds * ds + bias[f];
    out[i] = fmaxf(v, 0.0f);
}

// ---------------------------------------------------------------------------
// Host-side orchestration
// ---------------------------------------------------------------------------
extern "C" void kernel_launch(void* const* d_in, const int* in_sizes, int n_in,
                              void* d_out, int out_size, void* d_ws, size_t ws_size,
                              hipStream_t stream) {
    const float*     x    = (const float*)d_in[0];
    const long long* eidx = (const long long*)d_in[1];   // int64 [2, E]
    const float*     W1   = (const float*)d_in[2];
    const float*     b1   = (const float*)d_in[3];
    const float*     W2   = (const float*)d_in[4];
    const float*     b2   = (const float*)d_in[5];

    const int F0 = 64, F1 = 128, F2 = 64;
    const int N  = in_sizes[0] / F0;      // 50000 (multiple of 16)
    const int E  = in_sizes[1] / 2;       // 1600000
    const long long* srcE = eidx;
    const long long* dstE = eidx + E;

    // workspace carve-out
    char*  ws  = (char*)d_ws;
    int*   deg = (int*)ws;    ws += (size_t)N * sizeof(int);
    float* dis = (float*)ws;  ws += (size_t)N * sizeof(float);
    float* h1  = (float*)ws;  ws += (size_t)N * F1 * sizeof(float);
    float* a1  = (float*)ws;  ws += (size_t)N * F1 * sizeof(float);
    float* h2  = (float*)ws;  ws += (size_t)N * F2 * sizeof(float);
    float* a2  = (float*)ws;  ws += (size_t)N * F2 * sizeof(float);

    const int TB = 256;
    auto blocks = [](long long n, int tb) { return (unsigned)((n + tb - 1) / tb); };

    // zero accumulators (every call: harness does not re-poison between replays)
    gcn_zero_u32<<<blocks(N, TB), TB, 0, stream>>>((unsigned*)deg, N);
    gcn_zero_u32<<<blocks((long long)N * F1, TB), TB, 0, stream>>>((unsigned*)a1, (long long)N * F1);
    gcn_zero_u32<<<blocks((long long)N * F2, TB), TB, 0, stream>>>((unsigned*)a2, (long long)N * F2);

    // normalization terms
    gcn_degree<<<blocks(E, TB), TB, 0, stream>>>(dstE, deg, E);
    gcn_rsqrt<<<blocks(N, TB), TB, 0, stream>>>(deg, dis, N);

    // ---- layer 1: h1 = x @ W1 ; scatter ; relu(a1 + h1*dis^2 + b1) -> a1 ----
    {
        int tiles = (N >> 4) * (F1 >> 4);           // 3125 * 8
        gcn_gemm_wmma<<<blocks(tiles, 4) /*4 waves/block*/, 128, 0, stream>>>(
            x, W1, h1, N, F0, F1);
        long long work = (long long)E * (F1 >> 2);  // 51.2M threads
        gcn_scatter<<<blocks(work, TB), TB, 0, stream>>>(h1, srcE, dstE, dis, a1,
                                                         work, F1, 5);
        long long ne = (long long)N * F1;
        gcn_finish<<<blocks(ne, TB), TB, 0, stream>>>(a1, h1, dis, b1, a1, ne, F1, 7);
    }

    // ---- layer 2: h2 = a1 @ W2 ; scatter ; relu(a2 + h2*dis^2 + b2) -> out ----
    {
        int tiles = (N >> 4) * (F2 >> 4);           // 3125 * 4
        gcn_gemm_wmma<<<blocks(tiles, 4), 128, 0, stream>>>(a1, W2, h2, N, F1, F2);
        long long work = (long long)E * (F2 >> 2);  // 25.6M threads
        gcn_scatter<<<blocks(work, TB), TB, 0, stream>>>(h2, srcE, dstE, dis, a2,
                                                         work, F2, 4);
        long long ne = (long long)N * F2;
        gcn_finish<<<blocks(ne, TB), TB, 0, stream>>>(a2, h2, dis, b2, (float*)d_out,
                                                      ne, F2, 6);
    }
}